// RoiPoolingConv_57638461112771
// MI455X (gfx1250) — compile-verified
//
#include <hip/hip_runtime.h>
#include <hip/hip_bf16.h>

// ROI crop + bilinear resize (TF2 half-pixel-center semantics).
// img: (256,256,512) f32, rois: (1,300,4) i32 (x,y,w,h), out: (1,300,14,14,512) f32.
//
// Roofline: ~120 MB streamed output + ~480 MB gathered reads (image fits in the
// 192 MB L2) @ 23.3 TB/s -> ~25-30 us; compute is 0.24 GFLOP (negligible).
// Data movement uses the CDNA5 Tensor Data Mover in *gather mode*: one D# pulls
// the 4 needed pixel channel-vectors (rows of a conceptual 65536x512 tensor)
// into LDS, tracked by TENSORcnt. Output stores are non-temporal so the
// write-once result doesn't evict image lines from L2.

#define POOL     14
#define NUM_ROIS 300
#define IMG_H    256
#define IMG_W    256
#define IMG_C    512

typedef unsigned int       u32;
typedef unsigned long long u64;
typedef u32 __attribute__((ext_vector_type(4))) u32x4;
typedef int __attribute__((ext_vector_type(4))) i32x4;
typedef int __attribute__((ext_vector_type(8))) i32x8;

// Faithful port of _axis_coords: i1 computed from UNCLIPPED floor, frac from
// CLIPPED i0 (clips are no-ops for these input ranges but kept for fidelity).
__device__ __forceinline__ void axis_coords(int start, int size, int i, int limit,
                                            int& i0, int& i1, float& frac) {
    const float sizef = (float)size;
    float rel = ((float)i + 0.5f) * (sizef / (float)POOL) - 0.5f;
    rel = fminf(fmaxf(rel, 0.0f), sizef - 1.0f);
    const float absc = (float)start + rel;
    const int f = (int)floorf(absc);
    int j1 = min(f + 1, start + size - 1);
    int j0 = min(max(f, 0), limit - 1);
    j1 = min(max(j1, 0), limit - 1);
    i0 = j0;
    i1 = j1;
    frac = absc - (float)j0;
}

__device__ __forceinline__ float4 lerp4(float4 a, float4 b, float t) {
    float4 r;
    r.x = a.x * (1.0f - t) + b.x * t;
    r.y = a.y * (1.0f - t) + b.y * t;
    r.z = a.z * (1.0f - t) + b.z * t;
    r.w = a.w * (1.0f - t) + b.w * t;
    return r;
}

__global__ __launch_bounds__(128)
void roi_crop_resize_kernel(const float* __restrict__ img,
                            const int*   __restrict__ rois,
                            float*       __restrict__ out) {
    // LDS tile: 4 source pixels x 512 channels (8 KB)
    __shared__ float tile[4 * IMG_C];

    const int bid = blockIdx.x;                 // 0 .. 300*14*14-1
    const int roi = bid / (POOL * POOL);
    const int ij  = bid % (POOL * POOL);
    const int i   = ij / POOL;
    const int j   = ij % POOL;
    const int tid = threadIdx.x;                // 0..127, 4 channels each

    const int x = rois[roi * 4 + 0];
    const int y = rois[roi * 4 + 1];
    const int w = rois[roi * 4 + 2];
    const int h = rois[roi * 4 + 3];

    int r0, r1, c0, c1;
    float fr, fc;
    axis_coords(x, w, i, IMG_H, r0, r1, fr);    // rows
    axis_coords(y, h, j, IMG_W, c0, c1, fc);    // cols

    float4 v00, v01, v10, v11;

#if __has_builtin(__builtin_amdgcn_tensor_load_to_lds)
    // ---- CDNA5 TDM gather: 4 rows of a (65536 x 512) f32 tensor -> LDS ----
    if (tid < 32) {  // wave 0 issues the DMA (TDM ignores EXEC; one issue/wave)
        const u32 lds_base = (u32)(size_t)(void*)tile;   // low 32 bits = LDS offset
        const u64 ga       = (u64)(size_t)(const void*)img;

        // D# group 0: count=1, gather_mode=1, 32-bit indices, type=2
        u32x4 g0;
        g0.x = 0xC0000001u;                              // gather_mode|idx32|count=1
        g0.y = lds_base;                                 // lds_addr (bytes)
        g0.z = (u32)ga;                                  // global_addr[31:0]
        g0.w = ((u32)(ga >> 32) & 0x01FFFFFFu) | 0x80000000u; // addr[56:32] | type=2

        // D# group 1: data_size=4B, tensor_dim0=512, tensor_dim1=65536,
        // tile_dim0=512, tile_dim1=4 valid indices, dim0_stride=512
        i32x8 g1;
        g1[0] = 0x00020000;                 // workgroup_mask=0, data_size=2 (4B)
        g1[1] = (int)(512u << 16);          // tensor_dim0[15:0]=512 in bits 63:48
        g1[2] = 0;                          // tensor_dim0[31:16]=0, tensor_dim1 lo16=0
        g1[3] = (int)((512u << 16) | 1u);   // tensor_dim1 hi16=1 (65536), tile_dim0=512
        g1[4] = 4;                          // tile_dim1 = 4 gather rows
        g1[5] = 512;                        // tensor_dim0_stride[31:0]
        g1[6] = 0;
        g1[7] = 0;

        // Row indices (block-uniform -> force into SGPRs; folds to SALU since
        // the compiler scalarizes the ROI fetch via s_load_b128)
        i32x4 g2;
        g2[0] = __builtin_amdgcn_readfirstlane(r0 * IMG_W + c0);  // v00
        g2[1] = __builtin_amdgcn_readfirstlane(r0 * IMG_W + c1);  // v01
        g2[2] = __builtin_amdgcn_readfirstlane(r1 * IMG_W + c0);  // v10
        g2[3] = __builtin_amdgcn_readfirstlane(r1 * IMG_W + c1);  // v11
        i32x4 g3 = {0, 0, 0, 0};

#if defined(__clang_major__) && (__clang_major__ >= 23)
        i32x8 g4 = {0, 0, 0, 0, 0, 0, 0, 0};
        __builtin_amdgcn_tensor_load_to_lds(g0, g1, g2, g3, g4, 0);
#else
        __builtin_amdgcn_tensor_load_to_lds(g0, g1, g2, g3, 0);
#endif
#if __has_builtin(__builtin_amdgcn_s_wait_tensorcnt)
        __builtin_amdgcn_s_wait_tensorcnt(0);
#else
        asm volatile("s_wait_tensorcnt 0x0" ::: "memory");
#endif
    }
    __syncthreads();

    const int co = tid * 4;
    v00 = *(const float4*)&tile[0 * IMG_C + co];
    v01 = *(const float4*)&tile[1 * IMG_C + co];
    v10 = *(const float4*)&tile[2 * IMG_C + co];
    v11 = *(const float4*)&tile[3 * IMG_C + co];
#else
    // ---- Fallback: direct coalesced global_load_b128 gathers ----
    const int co = tid * 4;
    v00 = *(const float4*)(img + ((size_t)(r0 * IMG_W + c0)) * IMG_C + co);
    v01 = *(const float4*)(img + ((size_t)(r0 * IMG_W + c1)) * IMG_C + co);
    v10 = *(const float4*)(img + ((size_t)(r1 * IMG_W + c0)) * IMG_C + co);
    v11 = *(const float4*)(img + ((size_t)(r1 * IMG_W + c1)) * IMG_C + co);
#endif

    // Structurally identical to reference: column lerp, then row lerp.
    const float4 top = lerp4(v00, v01, fc);
    const float4 bot = lerp4(v10, v11, fc);
    const float4 res = lerp4(top, bot, fr);

    // Write-once output: non-temporal so it doesn't evict image lines from L2.
    float* dst = out + (size_t)bid * IMG_C + tid * 4;
#if __has_builtin(__builtin_nontemporal_store)
    __builtin_nontemporal_store(res.x, dst + 0);
    __builtin_nontemporal_store(res.y, dst + 1);
    __builtin_nontemporal_store(res.z, dst + 2);
    __builtin_nontemporal_store(res.w, dst + 3);
#else
    *(float4*)dst = res;
#endif
}

extern "C" void kernel_launch(void* const* d_in, const int* in_sizes, int n_in,
                              void* d_out, int out_size, void* d_ws, size_t ws_size,
                              hipStream_t stream) {
    const float* img  = (const float*)d_in[0];  // (256,256,512) f32
    const int*   rois = (const int*)d_in[1];    // (1,300,4) i32
    float*       out  = (float*)d_out;          // (1,300,14,14,512) f32

    const int blocks = NUM_ROIS * POOL * POOL;  // 58800
    roi_crop_resize_kernel<<<blocks, 128, 0, stream>>>(img, rois, out);
}